// LossDSAC_13099650253573
// MI455X (gfx1250) — compile-verified
//
#include <hip/hip_runtime.h>
#include <math.h>

// LossDSAC fused implementation for MI455X (gfx1250, wave32).
// Traffic ~21MB -> memory bound (~1us @ 23.3 TB/s); all heavy lifting fused
// into one image pass. Batched mean-reduction of d1 runs on the matrix unit
// via chained V_WMMA_F32_16X16X4_F32 (D = A x ones + C).

#define B_   16
#define M_   256
#define N_   256
#define L_   200
#define BMN_ (B_ * M_ * N_)

typedef __attribute__((ext_vector_type(2))) float v2f;
typedef __attribute__((ext_vector_type(8))) float v8f;

// ---------------------------------------------------------------------------
// Kernel A: one wave32. Stage d1 (snake & target) into LDS, then reduce the
// 16x200 matrices against a ones-vector with 100 chained f32 WMMAs.
// Result: ws_f[b] = (mean d1_snake - mean d1_target). Also zero stat counters.
// ---------------------------------------------------------------------------
__global__ void derivs_wmma_kernel(const float* __restrict__ snakes,
                                   const float* __restrict__ tsnake,
                                   float* __restrict__ ws_f,
                                   int* __restrict__ ws_i)
{
    __shared__ float d1s[B_][L_];
    __shared__ float d1t[B_][L_];
    const int tid = threadIdx.x;  // 32 threads, single wave, EXEC stays full

    // zero stats region ws_i[16..80)
    for (int i = tid; i < 64; i += 32) ws_i[16 + i] = 0;

    for (int idx = tid; idx < B_ * L_; idx += 32) {  // 3200/32 = uniform trips
        const int b  = idx / L_;
        const int l  = idx % L_;
        const int lp = (l == 0) ? (L_ - 1) : (l - 1);
        const int ln = (l == L_ - 1) ? 0 : (l + 1);
        const float* sb = snakes + (size_t)b * L_ * 2;
        float du = sb[ln * 2 + 0] - sb[lp * 2 + 0];
        float dv = sb[ln * 2 + 1] - sb[lp * 2 + 1];
        d1s[b][l] = sqrtf(du * du + dv * dv);
        const float* tb = tsnake + (size_t)b * L_ * 2;
        float tu = tb[ln * 2 + 0] - tb[lp * 2 + 0];
        float tv = tb[ln * 2 + 1] - tb[lp * 2 + 1];
        d1t[b][l] = sqrtf(tu * tu + tv * tv);
    }
    __syncthreads();

    // A-matrix 16x4 f32 layout: lanes 0-15 hold M=0..15; VGPR0={K0,K2},
    // VGPR1={K1,K3} -> lane gets K-pair (kh*2, kh*2+1), kh = lane>>4.
    const int m  = tid & 15;
    const int kh = (tid >> 4) & 1;
    v2f bones;  bones.x = 1.0f;  bones.y = 1.0f;  // ones matrix, layout-invariant
    v8f acc = {};
    for (int chunk = 0; chunk < L_ / 4; ++chunk) {  // +sum d1 (snake)
        const int k0 = chunk * 4 + kh * 2;
        v2f a;  a.x = d1s[m][k0];  a.y = d1s[m][k0 + 1];
        acc = __builtin_amdgcn_wmma_f32_16x16x4_f32(
            false, a, false, bones, (short)0, acc, false, false);
    }
    for (int chunk = 0; chunk < L_ / 4; ++chunk) {  // -sum d1 (target): F32 WMMA
        const int k0 = chunk * 4 + kh * 2;          // has no A-neg bit -> negate data
        v2f a;  a.x = -d1t[m][k0];  a.y = -d1t[m][k0 + 1];
        acc = __builtin_amdgcn_wmma_f32_16x16x4_f32(
            false, a, false, bones, (short)0, acc, false, false);
    }
    // C/D layout: lane n = tid%16, VGPR j holds M = j + 8*(tid/16).
    // Every column equals the row sum; read column 0 from lanes 0 and 16.
    if (tid == 0) {
        for (int j = 0; j < 8; ++j) ws_f[j] = acc[j] * (1.0f / (float)L_);
    } else if (tid == 16) {
        for (int j = 0; j < 8; ++j) ws_f[8 + j] = acc[j] * (1.0f / (float)L_);
    }
}

// ---------------------------------------------------------------------------
// Kernel B: one block per (b, row), 256 threads = columns.
//  Phase 1 (cooperative, per row): scanline x-intersections for the parity
//  fill; compacted lists of brush-gated vertices for snake/target draws.
//  Phase 2 (per pixel): parity count, brush max-gather, write all 4 gradient
//  images, LDS-reduce IoU/area stats -> global atomics.
// ---------------------------------------------------------------------------
__global__ void image_kernel(const float* __restrict__ snakes,
                             const float* __restrict__ tsnake,
                             const float* __restrict__ tmask,
                             const float* __restrict__ ws_f,
                             int* __restrict__ ws_i,
                             float* __restrict__ out)
{
    const int blk = blockIdx.x;
    const int b   = blk / M_;
    const int r   = blk % M_;
    const int tid = threadIdx.x;
    const float rf = (float)r;
    const float cf = (float)tid;

    const size_t pix = (size_t)b * (M_ * N_) + (size_t)r * N_ + tid;
    __builtin_prefetch(&tmask[pix], 0, 1);  // global_prefetch_b8

    __shared__ float xq[L_];                                   // fill x-ints
    __shared__ float lsV[L_], lsRV[L_], lsRU2[L_], lsD2[L_];   // snake draw list
    __shared__ float ltV[L_], ltRV[L_], ltRU2[L_], ltD2[L_];   // target draw list
    __shared__ int nS, nT;
    __shared__ int cInter, cUnion, cGt, cSnk;

    if (tid == 0) { nS = 0; nT = 0; cInter = 0; cUnion = 0; cGt = 0; cSnk = 0; }
    __syncthreads();

    const float* sb = snakes + (size_t)b * L_ * 2;
    const float* tb = tsnake + (size_t)b * L_ * 2;

    for (int e = tid; e < L_; e += blockDim.x) {
        // ---- polygon-fill precompute (depends on row only) ----
        const int e2 = (e == L_ - 1) ? 0 : e + 1;
        float u1 = sb[e * 2 + 0],  v1 = sb[e * 2 + 1];
        float u2 = sb[e2 * 2 + 0], v2 = sb[e2 * 2 + 1];
        bool straddle = (u1 > rf) != (u2 > rf);
        float denom = (u2 == u1) ? 1.0f : (u2 - u1);
        float xint = (v2 - v1) * (rf - u1) / denom + v1;
        xq[e] = straddle ? xint : -3.0e38f;   // never satisfies c < xq

        // ---- brush-draw row gate + compaction: snake ----
        const int lp = (e == 0) ? L_ - 1 : e - 1;
        const int ln = e2;
        if (fabsf(rf - rintf(u1)) <= 3.0f) {
            float du = sb[ln * 2 + 0] + sb[lp * 2 + 0] - 2.0f * u1;
            float dv = sb[ln * 2 + 1] + sb[lp * 2 + 1] - 2.0f * v1;
            float d2 = sqrtf(du * du + dv * dv);
            int i = atomicAdd(&nS, 1);
            float dr = rf - u1;
            lsV[i] = v1;  lsRV[i] = rintf(v1);  lsRU2[i] = dr * dr;  lsD2[i] = d2;
        }
        // ---- brush-draw row gate + compaction: target ----
        float tu = tb[e * 2 + 0], tv = tb[e * 2 + 1];
        if (fabsf(rf - rintf(tu)) <= 3.0f) {
            float du = tb[ln * 2 + 0] + tb[lp * 2 + 0] - 2.0f * tu;
            float dv = tb[ln * 2 + 1] + tb[lp * 2 + 1] - 2.0f * tv;
            float d2 = sqrtf(du * du + dv * dv);
            int i = atomicAdd(&nT, 1);
            float dr = rf - tu;
            ltV[i] = tv;  ltRV[i] = rintf(tv);  ltRU2[i] = dr * dr;  ltD2[i] = d2;
        }
    }
    __syncthreads();

    // ---- per-pixel parity fill ----
    int cnt = 0;
    for (int e = 0; e < L_; ++e) cnt += (cf < xq[e]) ? 1 : 0;
    const float mask = (float)(cnt & 1);

    // ---- per-pixel brush max-gather ----
    float hitS = 0.0f, mS = 0.0f;
    const int ns = nS;
    for (int i = 0; i < ns; ++i) {
        if (fabsf(cf - lsRV[i]) <= 3.0f) {
            float dc = cf - lsV[i];
            if (lsRU2[i] + dc * dc <= 4.0f) { hitS = 1.0f; mS = fmaxf(mS, lsD2[i]); }
        }
    }
    float hitT = 0.0f, mT = 0.0f;
    const int nt = nT;
    for (int i = 0; i < nt; ++i) {
        if (fabsf(cf - ltRV[i]) <= 3.0f) {
            float dc = cf - ltV[i];
            if (ltRU2[i] + dc * dc <= 4.0f) { hitT = 1.0f; mT = fmaxf(mT, ltD2[i]); }
        }
    }

    const float tgt = tmask[pix];
    out[80 + 0 * (size_t)BMN_ + pix] = hitS - hitT;   // g_edges
    out[80 + 1 * (size_t)BMN_ + pix] = ws_f[b];       // g_alpha (broadcast scalar)
    out[80 + 2 * (size_t)BMN_ + pix] = mS - mT;       // g_beta
    out[80 + 3 * (size_t)BMN_ + pix] = tgt - mask;    // g_kappa

    // ---- stats ----
    const float both = tgt + mask;
    if (both == 2.0f) atomicAdd(&cInter, 1);
    if (both >= 1.0f) atomicAdd(&cUnion, 1);
    if (tgt  > 0.0f)  atomicAdd(&cGt, 1);
    if (mask > 0.0f)  atomicAdd(&cSnk, 1);
    __syncthreads();
    if (tid == 0) {
        atomicAdd(&ws_i[16 + b], cInter);
        atomicAdd(&ws_i[32 + b], cUnion);
        atomicAdd(&ws_i[48 + b], cGt);
        atomicAdd(&ws_i[64 + b], cSnk);
    }
}

// ---------------------------------------------------------------------------
// Kernel C: finalize scalar outputs.
// ---------------------------------------------------------------------------
__global__ void finalize_kernel(const int* __restrict__ ws_i,
                                float* __restrict__ out)
{
    const int b = threadIdx.x;
    if (b >= B_) return;
    const float inter = (float)ws_i[16 + b];
    const float uni   = (float)ws_i[32 + b];
    const float size  = (float)(M_ * N_);
    out[ 0 + b] = inter / uni;          // iou
    out[16 + b] = inter / size;         // inter_frac
    out[32 + b] = uni / size;           // union_frac
    out[48 + b] = (float)ws_i[48 + b];  // area_gt
    out[64 + b] = (float)ws_i[64 + b];  // area_snake
}

extern "C" void kernel_launch(void* const* d_in, const int* in_sizes, int n_in,
                              void* d_out, int out_size, void* d_ws, size_t ws_size,
                              hipStream_t stream)
{
    // inputs: 0=edges 1=alpha 2=beta 3=kappa 4=snakes 5=target_mask 6=target_snake
    const float* snakes = (const float*)d_in[4];
    const float* tmask  = (const float*)d_in[5];
    const float* tsnake = (const float*)d_in[6];
    float* out  = (float*)d_out;
    float* ws_f = (float*)d_ws;
    int*   ws_i = (int*)d_ws;

    derivs_wmma_kernel<<<1, 32, 0, stream>>>(snakes, tsnake, ws_f, ws_i);
    image_kernel<<<B_ * M_, N_, 0, stream>>>(snakes, tsnake, tmask, ws_f, ws_i, out);
    finalize_kernel<<<1, 32, 0, stream>>>(ws_i, out);
}